// GHNN_layer_18184891531605
// MI455X (gfx1250) — compile-verified
//
#include <hip/hip_runtime.h>

// ---- problem constants (from reference) ------------------------------------
#define N_NODES 100000
#define N_EDGES 3200000
#define D_IN    256
#define D_OUT   256

// ---- ext-vector types for WMMA fragments -----------------------------------
typedef __attribute__((ext_vector_type(2))) float v2f;   // A/B frag 16x4 f32
typedef __attribute__((ext_vector_type(4))) float v4f;
typedef __attribute__((ext_vector_type(8))) float v8f;   // C/D frag 16x16 f32

// ---- GEMM tiling -----------------------------------------------------------
#define BM 128            // rows per block  (8 waves x 16 rows)
#define BN 64             // cols per block  (4 x 16-wide WMMA tiles per wave)
#define WL_STRIDE 160     // floats; 640B row stride -> hi-half lanes hit banks 32-63

// =============================================================================
// Kernel 1: support = X @ W  via V_WMMA_F32_16X16X4_F32 (full fp32 precision;
// the GEMM is memory-bound on MI455X so there is no reason to drop precision).
// Verified codegen: clustered global_load_b64 + ds_load_2addr_b64 + 16 WMMAs
// per unrolled iteration.
// =============================================================================
__global__ __launch_bounds__(256)
void gemm_xw_wmma(const float* __restrict__ X, const float* __restrict__ W,
                  float* __restrict__ S) {
  // W slice [256 x 64] staged pair-interleaved: WL[(k>>1)*STRIDE + n*2 + (k&1)]
  // so the per-lane B fragment {W[k][n], W[k+1][n]} is one contiguous b64 load.
  __shared__ float WL[(D_IN / 2) * WL_STRIDE];   // 80 KB of the 320 KB WGP LDS

  const int tid = threadIdx.x;
  const int nb  = blockIdx.y * BN;

  for (int idx = tid; idx < D_IN * BN; idx += 256) {
    const int k = idx >> 6;      // 0..255  (coalesced: consecutive tid -> consecutive n)
    const int n = idx & 63;
    WL[(k >> 1) * WL_STRIDE + n * 2 + (k & 1)] = W[k * D_OUT + nb + n];
  }
  __syncthreads();

  const int wave = tid >> 5;
  const int lane = tid & 31;
  const int hi   = lane >> 4;    // half-wave select (K pairs 0/1 vs 2/3)
  const int lo   = lane & 15;

  const int m_base = blockIdx.x * BM + wave * 16;
  int arow = m_base + lo;
  if (arow > N_NODES - 1) arow = N_NODES - 1;          // clamp (keeps EXEC all-1s)
  const float* aptr = X + (size_t)arow * D_IN + 2 * hi;

  v8f acc0 = {}, acc1 = {}, acc2 = {}, acc3 = {};

  #pragma unroll 4
  for (int s = 0; s < D_IN / 4; ++s) {
    // A fragment: lane(lo,hi) holds X[m_base+lo][4s+2hi .. 4s+2hi+1]
    v2f a = *(const v2f*)(aptr + 4 * s);
    // B fragments: lane(lo,hi) holds W[4s+2hi][n], W[4s+2hi+1][n]
    const float* wrow = &WL[(2 * s + hi) * WL_STRIDE + lo * 2];
    v2f b0 = *(const v2f*)(wrow + 0 * 32);
    v2f b1 = *(const v2f*)(wrow + 1 * 32);
    v2f b2 = *(const v2f*)(wrow + 2 * 32);
    v2f b3 = *(const v2f*)(wrow + 3 * 32);
    acc0 = __builtin_amdgcn_wmma_f32_16x16x4_f32(false, a, false, b0, (short)0, acc0, false, false);
    acc1 = __builtin_amdgcn_wmma_f32_16x16x4_f32(false, a, false, b1, (short)0, acc1, false, false);
    acc2 = __builtin_amdgcn_wmma_f32_16x16x4_f32(false, a, false, b2, (short)0, acc2, false, false);
    acc3 = __builtin_amdgcn_wmma_f32_16x16x4_f32(false, a, false, b3, (short)0, acc3, false, false);
  }

  // C/D layout: VGPR i -> row m_base + 8*hi + i, col nb + nt*16 + lo
  #pragma unroll
  for (int i = 0; i < 8; ++i) {
    const int r = m_base + hi * 8 + i;
    if (r < N_NODES) {
      float* o = S + (size_t)r * D_OUT + nb + lo;
      o[0]  = acc0[i];
      o[16] = acc1[i];
      o[32] = acc2[i];
      o[48] = acc3[i];
    }
  }
}

// =============================================================================
// Kernel 2: output[i][j] = bias[j]  (init before atomic scatter)
// =============================================================================
__global__ __launch_bounds__(256)
void bias_init(const float* __restrict__ bias, float* __restrict__ out) {
  const size_t idx = (size_t)blockIdx.x * 256 + threadIdx.x;   // grid covers N*256 exactly
  out[idx] = bias[idx & (D_OUT - 1)];
}

// Native fp32 atomic add, no return (STOREcnt-tracked), device scope so the
// RMW is coherent across workgroups. Guarantees GLOBAL_ATOMIC_ADD_F32 instead
// of a possible CAS-loop lowering of __hip_atomic_fetch_add.
__device__ __forceinline__ void atomic_add_f32_dev(float* p, float v) {
  asm volatile("global_atomic_add_f32 %0, %1, off scope:SCOPE_DEV"
               :: "v"(p), "v"(v)
               : "memory");
}

// =============================================================================
// Kernel 3: COO SpMM scatter. One wave per edge: 32 lanes x 8 floats = 256-wide
// row. support (102 MB) and output (102 MB) both fit in the 192 MB L2, so the
// gathers and fp32 atomics stay on-chip. Edge indices are wave-uniform ->
// readfirstlane them so row base addresses are scalar (SADDR addressing).
// =============================================================================
__global__ __launch_bounds__(256)
void spmm_scatter(const float* __restrict__ S, const float* __restrict__ ev,
                  const int* __restrict__ er, const int* __restrict__ ec,
                  float* __restrict__ out) {
  const int e = blockIdx.x * 8 + (threadIdx.x >> 5);   // one edge per wave32
  const int lane = threadIdx.x & 31;

  const int   r = __builtin_amdgcn_readfirstlane(er[e]);
  const int   c = __builtin_amdgcn_readfirstlane(ec[e]);
  const float v = ev[e];

  const v4f* src = (const v4f*)(S + (size_t)c * D_OUT + lane * 8);
  const v4f s0 = src[0];
  const v4f s1 = src[1];

  float* dst = out + (size_t)r * D_OUT + lane * 8;
  #pragma unroll
  for (int i = 0; i < 4; ++i) {
    atomic_add_f32_dev(dst + i,     v * s0[i]);
    atomic_add_f32_dev(dst + 4 + i, v * s1[i]);
  }
}

// =============================================================================
extern "C" void kernel_launch(void* const* d_in, const int* in_sizes, int n_in,
                              void* d_out, int out_size, void* d_ws, size_t ws_size,
                              hipStream_t stream) {
  const float* X    = (const float*)d_in[0];   // input_feature [100000,256] f32
  const float* W    = (const float*)d_in[1];   // weight        [256,256]    f32
  const float* bias = (const float*)d_in[2];   // bias          [256]        f32
  const float* ev   = (const float*)d_in[3];   // edge_val      [3.2M]       f32
  const int*   er   = (const int*)d_in[4];     // edge_row      [3.2M]       i32
  const int*   ec   = (const int*)d_in[5];     // edge_col      [3.2M]       i32
  float*       out  = (float*)d_out;           // [100000,256] f32
  float*       S    = (float*)d_ws;            // support scratch: 102.4 MB

  dim3 g1((N_NODES + BM - 1) / BM, D_OUT / BN);        // 782 x 4
  gemm_xw_wmma<<<g1, 256, 0, stream>>>(X, W, S);

  bias_init<<<N_NODES, 256, 0, stream>>>(bias, out);   // 100000 * 256 elems

  // N_EDGES is divisible by 8, so every block is fully populated (no guard
  // needed -> uniform control flow).
  spmm_scatter<<<N_EDGES / 8, 256, 0, stream>>>(S, ev, er, ec, out);
}